// GaussianSoftBoundaryLoss_12446815224157
// MI455X (gfx1250) — compile-verified
//
#include <hip/hip_runtime.h>
#include <stdint.h>
#include <math.h>

// GaussianSoftBoundaryLoss for MI455X (gfx1250, wave32).
// B=512 rows, T=16384 cols. Memory-bound: 48MB in -> ~2.2us at 23.3TB/s.
// Strategy: 1 block per row; async-load bools to LDS (ASYNCcnt path);
// register bitmask distance scans; integer-distance Gaussian via LDS LUT;
// TRANS-unit exp/log for the BCE term; WMMA f32 16x16x4 final reduction.

#define T_LEN   16384
#define TPB     256
#define CHUNK   64          // elements per thread (TPB*CHUNK == T_LEN)
#define SSTRIDE 80          // padded LDS slot stride (bytes) for bool staging
#define NEGI    (-(1 << 20))
#define POSI    ((1 << 20))

typedef float v2f __attribute__((ext_vector_type(2)));
typedef float v8f __attribute__((ext_vector_type(8)));
typedef int v4i_vs __attribute__((vector_size(16)));
typedef __attribute__((address_space(1))) v4i_vs* gas_v4p;  // global (device) AS
typedef __attribute__((address_space(3))) v4i_vs* las_v4p;  // LDS (shared) AS

__device__ __forceinline__ int imin(int a, int b) { return a < b ? a : b; }
__device__ __forceinline__ int imax(int a, int b) { return a > b ? a : b; }

// Async copy 16 bytes global -> LDS (CDNA5 GLOBAL_LOAD_ASYNC_TO_LDS_B128).
__device__ __forceinline__ void async_ld16(const void* g, void* l) {
  unsigned long long ga = (unsigned long long)(uintptr_t)g;
  unsigned la = (unsigned)(uintptr_t)l;  // generic LDS ptr low 32b == LDS addr
#if __has_builtin(__builtin_amdgcn_global_load_async_to_lds_b128)
  __builtin_amdgcn_global_load_async_to_lds_b128((gas_v4p)ga, (las_v4p)la, 0, 0);
#else
  asm volatile("global_load_async_to_lds_b128 %0, %1, off"
               :: "v"(la), "v"(ga) : "memory");
#endif
}

__device__ __forceinline__ void wait_async0() {
#if __has_builtin(__builtin_amdgcn_s_wait_asynccnt)
  __builtin_amdgcn_s_wait_asynccnt(0);
#else
  asm volatile("s_wait_asynccnt 0x0" ::: "memory");
#endif
  asm volatile("" ::: "memory");
}

// Turn 4 bool bytes (0/1) in a u32 into 4 bits.
__device__ __forceinline__ unsigned nyb(unsigned w) {
  w &= 0x01010101u;
  return (w | (w >> 7) | (w >> 14) | (w >> 21)) & 0xFu;
}
__device__ __forceinline__ unsigned bits16(uint4 v) {
  return nyb(v.x) | (nyb(v.y) << 4) | (nyb(v.z) << 8) | (nyb(v.w) << 12);
}

__global__ __launch_bounds__(TPB) void gsbl_main(
    const float* __restrict__ logits,
    const unsigned char* __restrict__ boundaries,
    const unsigned char* __restrict__ mask,
    float* __restrict__ wsLoss,
    float* __restrict__ wsCnt) {
  __shared__ float lut[256];
  __shared__ int   sLast[TPB];
  __shared__ int   sFirst[TPB];
  __shared__ float redA[TPB];
  __shared__ float redB[TPB];
  __shared__ unsigned char stageB[TPB * SSTRIDE];
  __shared__ unsigned char stageM[TPB * SSTRIDE];

  const int row = blockIdx.x;
  const int t = threadIdx.x;
  const size_t rb = (size_t)row * T_LEN;

  // ---- Stage this thread's 64 bool bytes of each array into LDS (async) ----
  const unsigned char* gB = boundaries + rb + (size_t)t * CHUNK;
  const unsigned char* gM = mask + rb + (size_t)t * CHUNK;
  unsigned char* lB = &stageB[t * SSTRIDE];
  unsigned char* lM = &stageM[t * SSTRIDE];
#pragma unroll
  for (int k = 0; k < 4; ++k) {
    async_ld16(gB + k * 16, lB + k * 16);
    async_ld16(gM + k * 16, lM + k * 16);
  }

  // Gaussian LUT over integer distances (overlaps with the async loads).
  // exp(-0.5*d^2) underflows to 0 for d>=15, so clamping d at 255 is exact.
  lut[t] = expf(-0.5f * (float)(t * t));

  wait_async0();  // own wave's asyncs done -> own LDS slot valid

  // ---- Build 64-bit masks: mb = boundaries&mask, mm = mask ----
  unsigned long long mb = 0ull, mm = 0ull;
#pragma unroll
  for (int k = 0; k < 4; ++k) {
    uint4 vb = ((const uint4*)lB)[k];
    uint4 vm = ((const uint4*)lM)[k];
    unsigned bbits = bits16(vb);
    unsigned mbits = bits16(vm);
    mb |= (unsigned long long)(bbits & mbits) << (k * 16);
    mm |= (unsigned long long)mbits << (k * 16);
  }

  const int base_i = t * CHUNK;
  sLast[t]  = mb ? (base_i + 63 - __clzll(mb)) : NEGI;   // last boundary in chunk
  sFirst[t] = mb ? (base_i + __ffsll(mb) - 1) : POSI;    // first boundary in chunk
  __syncthreads();

  // ---- Hillis-Steele inclusive cummax(sLast) / reverse cummin(sFirst) ----
#pragma unroll
  for (int off = 1; off < TPB; off <<= 1) {
    int myL = sLast[t], myF = sFirst[t];
    int nbL = (t >= off) ? sLast[t - off] : NEGI;
    int nbF = (t + off < TPB) ? sFirst[t + off] : POSI;
    __syncthreads();
    sLast[t] = imax(myL, nbL);
    sFirst[t] = imin(myF, nbF);
    __syncthreads();
  }
  const int prefLast = (t > 0) ? sLast[t - 1] : NEGI;        // exclusive prefix max
  const int sufFirst = (t < TPB - 1) ? sFirst[t + 1] : POSI; // exclusive suffix min

  // ---- Forward sweep: left-distance per element, clamped to 255, packed u8 ----
  unsigned dlp[CHUNK / 4];
  {
    int last = prefLast;
#pragma unroll
    for (int k = 0; k < CHUNK / 4; ++k) {
      unsigned pk = 0;
#pragma unroll
      for (int j = 0; j < 4; ++j) {
        const int bit = k * 4 + j;
        const int i = base_i + bit;
        if ((mb >> bit) & 1ull) last = i;
        pk |= (unsigned)imin(i - last, 255) << (8 * j);
      }
      dlp[k] = pk;
    }
  }

  // ---- Backward sweep: right-distance, target LUT, masked BCE terms ----
  // BCE constant term max(x,0)+log1p(exp(-|x|)) computed with hardware TRANS
  // ops (v_exp_f32 / v_log_f32) that co-execute with VALU, keeping the kernel
  // HBM-bound instead of transcendental-bound.
  float base_sum = 0.0f, sub_sum = 0.0f;
  const float cnt = (float)__popcll(mm);
  {
    int nxt = sufFirst;
    const float4* Lrow = (const float4*)(logits + rb + (size_t)base_i);
#pragma unroll
    for (int k = CHUNK / 4 - 1; k >= 0; --k) {
      const float4 x4 = Lrow[k];
      const float xs[4] = {x4.x, x4.y, x4.z, x4.w};
      const unsigned pk = dlp[k];
#pragma unroll
      for (int j = 3; j >= 0; --j) {
        const int bit = k * 4 + j;
        const int i = base_i + bit;
        if ((mb >> bit) & 1ull) nxt = i;
        const int dr = imin(nxt - i, 255);
        const int dl = (int)((pk >> (8 * j)) & 0xFFu);
        const int d = imin(dl, dr);
        if ((mm >> bit) & 1ull) {
          const float x = xs[j];
          base_sum += fmaxf(x, 0.0f) + __logf(1.0f + __expf(-fabsf(x)));
          sub_sum += x * lut[d];
        }
      }
    }
  }

  // ---- Block reduction of (loss, mask-count) ----
  redA[t] = base_sum - sub_sum;
  redB[t] = cnt;
  __syncthreads();
#pragma unroll
  for (int s = TPB / 2; s > 0; s >>= 1) {
    if (t < s) {
      redA[t] += redA[t + s];
      redB[t] += redB[t + s];
    }
    __syncthreads();
  }
  if (t == 0) {
    wsLoss[row] = redA[0];
    wsCnt[row] = redB[0];
  }
}

// One wave32 reduces the per-row partials using V_WMMA_F32_16X16X4_F32 with
// A == ones: D[m,n] += sum_k B[k,n], so the grand total is layout-agnostic
// (sum of c[0] over all 32 lanes counts every column sum twice).
// Requires nrows % 64 == 0 (true: 512) -> no bounds checks, EXEC stays all-1s.
__global__ __launch_bounds__(32) void gsbl_finalize(
    const float* __restrict__ wsLoss, const float* __restrict__ wsCnt,
    float* __restrict__ out, int nrows) {
  const int l = threadIdx.x;
  v2f ones;
  ones[0] = 1.0f;
  ones[1] = 1.0f;
  v8f cL = {0.f, 0.f, 0.f, 0.f, 0.f, 0.f, 0.f, 0.f};
  v8f cC = {0.f, 0.f, 0.f, 0.f, 0.f, 0.f, 0.f, 0.f};
  for (int base = 0; base < nrows; base += 64) {
    v2f bL, bC;
    bL[0] = wsLoss[base + l];
    bL[1] = wsLoss[base + 32 + l];
    bC[0] = wsCnt[base + l];
    bC[1] = wsCnt[base + 32 + l];
    cL = __builtin_amdgcn_wmma_f32_16x16x4_f32(false, ones, false, bL, (short)0, cL, false, false);
    cC = __builtin_amdgcn_wmma_f32_16x16x4_f32(false, ones, false, bC, (short)0, cC, false, false);
  }
  float sL = cL[0], sC = cC[0];
#pragma unroll
  for (int m = 16; m >= 1; m >>= 1) {
    sL += __shfl_xor(sL, m, 32);
    sC += __shfl_xor(sC, m, 32);
  }
  sL *= 0.5f;  // every column sum appears in rows M=0 and M=8 of c[0]
  sC *= 0.5f;
  if (l == 0) out[0] = sL / (sC + 1e-6f);
}

extern "C" void kernel_launch(void* const* d_in, const int* in_sizes, int n_in,
                              void* d_out, int out_size, void* d_ws, size_t ws_size,
                              hipStream_t stream) {
  const float* logits = (const float*)d_in[0];
  const unsigned char* boundaries = (const unsigned char*)d_in[1];  // jnp bool = 1 byte
  const unsigned char* mask = (const unsigned char*)d_in[2];
  float* out = (float*)d_out;

  const int nrows = in_sizes[0] / T_LEN;  // 512 (multiple of 64)
  float* wsLoss = (float*)d_ws;
  float* wsCnt = wsLoss + nrows;

  gsbl_main<<<nrows, TPB, 0, stream>>>(logits, boundaries, mask, wsLoss, wsCnt);
  gsbl_finalize<<<1, 32, 0, stream>>>(wsLoss, wsCnt, out, nrows);
}